// GraphConv_75668733821114
// MI455X (gfx1250) — compile-verified
//
#include <hip/hip_runtime.h>

#define E_TOTAL 320000
#define NNODES  10000
#define D       256        // D_IN == D_OUT == K == N
#define M_TILE  64         // edges per block
#define KC      32         // K rows of W staged per panel
#define NPANEL  (D / KC)   // 8 panels
#define ASTRIDE 260        // 256 + 4 -> A rows 4 banks apart
#define WSTRIDE 272        // 256 + 16 -> W rows 16 banks apart

#define AS1 __attribute__((address_space(1)))
#define AS3 __attribute__((address_space(3)))

typedef __attribute__((ext_vector_type(8))) float v8f;
typedef __attribute__((ext_vector_type(2))) float v2f;
typedef __attribute__((ext_vector_type(4))) int   v4i;

#if __has_builtin(__builtin_amdgcn_global_load_async_to_lds_b128) && \
    __has_builtin(__builtin_amdgcn_global_store_async_from_lds_b128)
#define HAVE_ASYNC_LDS 1
#else
#define HAVE_ASYNC_LDS 0
#endif

__device__ __forceinline__ void async_wait0() {
#if HAVE_ASYNC_LDS
#if __has_builtin(__builtin_amdgcn_s_wait_asynccnt)
    __builtin_amdgcn_s_wait_asynccnt(0);
#else
    asm volatile("s_wait_asynccnt 0x0" ::: "memory");
#endif
#endif
}

__global__ __launch_bounds__(256)
void graphconv_wmma_f32(const float* __restrict__ x,
                        const long long* __restrict__ ei,
                        const float* __restrict__ W,
                        const float* __restrict__ bias,
                        float* __restrict__ out)
{
    __shared__ float As[M_TILE * ASTRIDE];       // 66560 B: A tile, later C tile
    __shared__ float Wp[2][KC * WSTRIDE];        // 2 x 34816 B: double-buffered W panel
    __shared__ int   rowIdx[M_TILE];
    __shared__ int   colIdx[M_TILE];

    const int tid  = threadIdx.x;
    const int wave = tid >> 5;
    const int lane = tid & 31;
    const int e0   = blockIdx.x * M_TILE;

    // cooperative copy coordinates (used by staging, gather, epilogue)
    const int c4v = tid & 63;   // float4 column index 0..63
    const int eb  = tid >> 6;   // 0..3

    // ---- per-wave output tile: 2 M-tiles x 4 N-subtiles (32 rows x 64 cols) ----
    const int mw    = wave & 1;          // 0..1
    const int nw    = wave >> 1;         // 0..3
    const int m0    = mw * 32;
    const int nbase = nw * 64;
    const int lhalf = lane & 15;         // row within half-wave
    const int khalf = (lane >> 4) * 2;   // K offset 0 or 2 (A/B frag layout)

    // ---- kick off panel-0 W staging as early as possible ----
#if HAVE_ASYNC_LDS
#pragma unroll
    for (int it = 0; it < 8; ++it) {
        const int i  = tid + 256 * it;
        const int kr = i >> 6;
        const int c4 = i & 63;
        __builtin_amdgcn_global_load_async_to_lds_b128(
            (AS1 v4i*)(W + (size_t)kr * D + c4 * 4),
            (AS3 v4i*)(&Wp[0][kr * WSTRIDE + c4 * 4]), 0, 0);
    }
#endif

    // ---- stage clipped edge indices (int64 -> int32) ----
    if (tid < M_TILE) {
        long long r = ei[e0 + tid];
        long long c = ei[(long long)E_TOTAL + e0 + tid];
        r = r < 0 ? 0 : (r > NNODES - 1 ? NNODES - 1 : r);
        c = c < 0 ? 0 : (c > NNODES - 1 ? NNODES - 1 : c);
        rowIdx[tid] = (int)r;
        colIdx[tid] = (int)c;
    }

    // per-lane bias values for the 4 N-subtiles this wave owns
    float bval[4];
#pragma unroll
    for (int j = 0; j < 4; ++j)
        bval[j] = bias[nbase + j * 16 + lhalf];

    __syncthreads();

    // ---- gather x[row] + x[col] into A tile (coalesced float4) ----
    for (int e = eb; e < M_TILE; e += 4) {
        const float4 xr = ((const float4*)(x + (size_t)rowIdx[e] * D))[c4v];
        const float4 xc = ((const float4*)(x + (size_t)colIdx[e] * D))[c4v];
        float4 s;
        s.x = xr.x + xc.x; s.y = xr.y + xc.y;
        s.z = xr.z + xc.z; s.w = xr.w + xc.w;
        *(float4*)(&As[e * ASTRIDE + c4v * 4]) = s;
    }

#if !HAVE_ASYNC_LDS
    // fallback: synchronous panel-0 staging
#pragma unroll
    for (int it = 0; it < 8; ++it) {
        const int i  = tid + 256 * it;
        const int kr = i >> 6;
        const int c4 = i & 63;
        const float4 w4 = ((const float4*)(W + (size_t)kr * D))[c4];
        *(float4*)(&Wp[0][kr * WSTRIDE + c4 * 4]) = w4;
    }
#else
    async_wait0();
#endif
    __syncthreads();

    v8f acc[2][4];
#pragma unroll
    for (int mi = 0; mi < 2; ++mi)
#pragma unroll
        for (int j = 0; j < 4; ++j)
#pragma unroll
            for (int i = 0; i < 8; ++i)
                acc[mi][j][i] = 0.0f;

    for (int p = 0; p < NPANEL; ++p) {
        const int buf = p & 1;
        const bool havenext = (p + 1) < NPANEL;

#if HAVE_ASYNC_LDS
        // DMA next panel into the other buffer; overlaps with WMMA block below
        if (havenext) {
#pragma unroll
            for (int it = 0; it < 8; ++it) {
                const int i  = tid + 256 * it;
                const int kr = i >> 6;
                const int c4 = i & 63;
                __builtin_amdgcn_global_load_async_to_lds_b128(
                    (AS1 v4i*)(W + (size_t)((p + 1) * KC + kr) * D + c4 * 4),
                    (AS3 v4i*)(&Wp[buf ^ 1][kr * WSTRIDE + c4 * 4]), 0, 0);
            }
        }
#else
        float4 pre[8];
        if (havenext) {
#pragma unroll
            for (int it = 0; it < 8; ++it) {
                const int i  = tid + 256 * it;
                const int kr = i >> 6;
                const int c4 = i & 63;
                pre[it] = ((const float4*)(W + (size_t)((p + 1) * KC + kr) * D))[c4];
            }
        }
#endif

        // compute on current panel
#pragma unroll
        for (int kk = 0; kk < KC; kk += 4) {
            const int kA = p * KC + kk + khalf;
            const v2f a0 = *(const v2f*)(&As[(m0 + lhalf)      * ASTRIDE + kA]);
            const v2f a1 = *(const v2f*)(&As[(m0 + 16 + lhalf) * ASTRIDE + kA]);
            v2f bf[4];
#pragma unroll
            for (int j = 0; j < 4; ++j) {
                const int n = nbase + j * 16 + lhalf;
                bf[j].x = Wp[buf][(kk + khalf)     * WSTRIDE + n];
                bf[j].y = Wp[buf][(kk + khalf + 1) * WSTRIDE + n];
            }
#pragma unroll
            for (int j = 0; j < 4; ++j) {
                acc[0][j] = __builtin_amdgcn_wmma_f32_16x16x4_f32(
                    false, a0, false, bf[j], (short)0, acc[0][j], false, false);
                acc[1][j] = __builtin_amdgcn_wmma_f32_16x16x4_f32(
                    false, a1, false, bf[j], (short)0, acc[1][j], false, false);
            }
        }

#if HAVE_ASYNC_LDS
        if (havenext) async_wait0();
#else
        if (havenext) {
#pragma unroll
            for (int it = 0; it < 8; ++it) {
                const int i  = tid + 256 * it;
                const int kr = i >> 6;
                const int c4 = i & 63;
                *(float4*)(&Wp[buf ^ 1][kr * WSTRIDE + c4 * 4]) = pre[it];
            }
        }
#endif
        __syncthreads();
    }

    // ---- write accumulators + bias to LDS (C/D layout) ----
    {
        const int rofs = (lane >> 4) * 8;  // lanes 16-31 hold M=8..15
#pragma unroll
        for (int mi = 0; mi < 2; ++mi)
#pragma unroll
            for (int j = 0; j < 4; ++j) {
                const int n = nbase + j * 16 + lhalf;
#pragma unroll
                for (int r = 0; r < 8; ++r)
                    As[(m0 + mi * 16 + rofs + r) * ASTRIDE + n] = acc[mi][j][r] + bval[j];
            }
    }
    __syncthreads();

    // ---- coalesced store of the C tile ----
#if HAVE_ASYNC_LDS
#pragma unroll
    for (int it = 0; it < 16; ++it) {
        const int e = eb + 4 * it;
        __builtin_amdgcn_global_store_async_from_lds_b128(
            (AS1 v4i*)(out + (size_t)(e0 + e) * D + c4v * 4),
            (AS3 v4i*)(&As[e * ASTRIDE + c4v * 4]), 0, 0);
    }
    async_wait0();
#else
    for (int e = eb; e < M_TILE; e += 4) {
        const float4 v = *(const float4*)(&As[e * ASTRIDE + c4v * 4]);
        ((float4*)(out + (size_t)(e0 + e) * D))[c4v] = v;
    }
#endif
}

extern "C" void kernel_launch(void* const* d_in, const int* in_sizes, int n_in,
                              void* d_out, int out_size, void* d_ws, size_t ws_size,
                              hipStream_t stream) {
    const float*      x    = (const float*)d_in[0];
    const long long*  ei   = (const long long*)d_in[1];
    const float*      W    = (const float*)d_in[2];
    const float*      bias = (const float*)d_in[3];
    float*            out  = (float*)d_out;

    const int blocks = E_TOTAL / M_TILE;   // 5000
    graphconv_wmma_f32<<<blocks, 256, 0, stream>>>(x, ei, W, bias, out);
}